// DSSIM_46849503265169
// MI455X (gfx1250) — compile-verified
//
#include <hip/hip_runtime.h>

typedef float v2f __attribute__((ext_vector_type(2)));
typedef float v8f __attribute__((ext_vector_type(8)));

#define IMG_H 512
#define IMG_W 512
#define TILE 32
#define HALO 5
#define IN_ROWS 48          // 3x16 row tiles covering the 42 needed rows
#define IN_COLS 44          // 32 + 2*5 halo, padded to 44
#define SSIM_C1 1.0e-4f     // 0.01^2
#define SSIM_C2 9.0e-4f     // 0.03^2
#define N_PIX 25165824.0f   // 32*3*512*512
#define N_BLOCKS (96 * 256) // planes * (512/32)^2

__global__ __launch_bounds__(128) void dssim_tile_kernel(
    const float* __restrict__ img1, const float* __restrict__ img2,
    const float* __restrict__ gk, float* __restrict__ partial) {
  __shared__ float sG[16];                       // 1D gaussian (11 taps)
  __shared__ float sF[5][IN_ROWS * IN_COLS];     // field planes w/ halo
  __shared__ float sRB[5][IN_ROWS * TILE];       // row-blurred fields
  __shared__ float sBL[5][TILE * TILE];          // fully blurred fields
  __shared__ float sWsum[4];

  const int tid  = threadIdx.x;
  const int lane = tid & 31;
  // Wave id as a *scalar* (SGPR): guards below become s_cmp/s_cbranch with
  // EXEC untouched, instead of saveexec chains around every WMMA block.
  const int wave = __builtin_amdgcn_readfirstlane(tid >> 5);
  const int bid  = blockIdx.x;
  const int plane   = bid >> 8;    // b*3 + c, 0..95
  const int tileIdx = bid & 255;
  const int iy0 = (tileIdx >> 4) * TILE - HALO;
  const int ix0 = (tileIdx & 15) * TILE - HALO;

  const float* __restrict__ p1 = img1 + (size_t)plane * (IMG_H * IMG_W);
  const float* __restrict__ p2 = img2 + (size_t)plane * (IMG_H * IMG_W);

  // Recover exact separable 1D kernel: g[i] = sum_j k2d[i][j]  (sum(g)=1)
  if (tid < 11) {
    const float* krow = gk + (size_t)(plane % 3) * 121 + tid * 11;
    float s = 0.f;
#pragma unroll
    for (int j = 0; j < 11; ++j) s += krow[j];
    sG[tid] = s;
  }

  // Load halo tiles and expand to the 5 field planes (zero-pad at borders).
  for (int i = tid; i < IN_ROWS * IN_COLS; i += 128) {
    int r = i / IN_COLS, c = i - r * IN_COLS;
    int gy = iy0 + r, gx = ix0 + c;
    bool ok = ((unsigned)gy < IMG_H) && ((unsigned)gx < IMG_W);
    size_t off = (size_t)gy * IMG_W + gx;
    float u = ok ? p1[off] : 0.f;
    float w = ok ? p2[off] : 0.f;
    sF[0][i] = u;
    sF[1][i] = w;
    sF[2][i] = u * u;
    sF[3][i] = w * w;
    sF[4][i] = u * w;
  }
  __syncthreads();

  const int M = lane & 15;   // A row index / D column index
  const int h = lane >> 4;   // half-wave selector (K pairs / D row halves)
  const int N = M;           // B column index

  // Task-invariant Toeplitz fragments, held in registers.
  // Row pass B operand:  B[k][c] = g[k - c]   (k = kk*4 + 2h (+1))
  // Col pass A operand:  A[r][k] = g[k - r]
  v2f bRow[7], aCol[7];
#pragma unroll
  for (int kk = 0; kk < 7; ++kk) {
    int k0 = kk * 4 + 2 * h;
    int d0 = k0 - N, d1 = k0 + 1 - N;
    bRow[kk].x = ((unsigned)d0 <= 10u) ? sG[d0] : 0.f;
    bRow[kk].y = ((unsigned)d1 <= 10u) ? sG[d1] : 0.f;
    int e0 = k0 - M, e1 = k0 + 1 - M;
    aCol[kk].x = ((unsigned)e0 <= 10u) ? sG[e0] : 0.f;
    aCol[kk].y = ((unsigned)e1 <= 10u) ? sG[e1] : 0.f;
  }

  // ---------- Row pass: rb = field x Toeplitz(g) via V_WMMA_F32_16X16X4_F32
  // 5 fields x (3 row-tiles x 2 col-tiles) = 30 tasks, compile-time unrolled,
  // scalar round-robin guard (wave is SGPR) keeps EXEC all-ones at every WMMA.
#pragma unroll
  for (int t = 0; t < 30; ++t) {
    if ((t & 3) == wave) {
      const int f = t / 6, s = t % 6, m = s >> 1, n = s & 1;
      v8f acc = {};
#pragma unroll
      for (int kk = 0; kk < 7; ++kk) {           // K = 28 (26 used)
        int k0 = kk * 4 + 2 * h;
        int rr = (16 * m + M) * IN_COLS + 16 * n + k0;
        v2f a;
        a.x = sF[f][rr];
        a.y = sF[f][rr + 1];
        acc = __builtin_amdgcn_wmma_f32_16x16x4_f32(false, a, false, bRow[kk],
                                                    (short)0, acc, false, false);
      }
#pragma unroll
      for (int i = 0; i < 8; ++i)
        sRB[f][(16 * m + i + 8 * h) * TILE + 16 * n + N] = acc[i];
    }
  }
  __syncthreads();

  // ---------- Column pass: blur = Toeplitz(g) x rb
  // 5 fields x (2x2 tiles) = 20 tasks.
#pragma unroll
  for (int t = 0; t < 20; ++t) {
    if ((t & 3) == wave) {
      const int f = t / 4, s = t % 4, m = s >> 1, n = s & 1;
      v8f acc = {};
#pragma unroll
      for (int kk = 0; kk < 7; ++kk) {
        int k0 = kk * 4 + 2 * h;
        v2f b;
        b.x = sRB[f][(16 * m + k0) * TILE + 16 * n + N];
        b.y = sRB[f][(16 * m + k0 + 1) * TILE + 16 * n + N];
        acc = __builtin_amdgcn_wmma_f32_16x16x4_f32(false, aCol[kk], false, b,
                                                    (short)0, acc, false, false);
      }
#pragma unroll
      for (int i = 0; i < 8; ++i)
        sBL[f][(16 * m + i + 8 * h) * TILE + 16 * n + N] = acc[i];
    }
  }
  __syncthreads();

  // ---------- Per-pixel SSIM + block reduction
  float sum = 0.f;
#pragma unroll
  for (int j = 0; j < 8; ++j) {
    int i = tid + j * 128;
    float mu1 = sBL[0][i], mu2 = sBL[1][i];
    float e11 = sBL[2][i], e22 = sBL[3][i], e12 = sBL[4][i];
    float mu1sq = mu1 * mu1, mu2sq = mu2 * mu2, mu12 = mu1 * mu2;
    float s11 = e11 - mu1sq, s22 = e22 - mu2sq, s12 = e12 - mu12;
    float num = (2.f * mu12 + SSIM_C1) * (2.f * s12 + SSIM_C2);
    float den = (mu1sq + mu2sq + SSIM_C1) * (s11 + s22 + SSIM_C2);
    sum += (1.f - num / den) * 0.5f;
  }
#pragma unroll
  for (int o = 16; o > 0; o >>= 1) sum += __shfl_xor(sum, o, 32);
  if (lane == 0) sWsum[wave] = sum;
  __syncthreads();
  if (tid == 0) partial[bid] = sWsum[0] + sWsum[1] + sWsum[2] + sWsum[3];
}

__global__ __launch_bounds__(256) void dssim_reduce_kernel(
    const float* __restrict__ partial, int n, float* __restrict__ out) {
  __shared__ float sW[8];
  float sum = 0.f;
  for (int i = threadIdx.x; i < n; i += 256) sum += partial[i];
#pragma unroll
  for (int o = 16; o > 0; o >>= 1) sum += __shfl_xor(sum, o, 32);
  int lane = threadIdx.x & 31, wave = threadIdx.x >> 5;
  if (lane == 0) sW[wave] = sum;
  __syncthreads();
  if (threadIdx.x == 0) {
    float tot = 0.f;
#pragma unroll
    for (int i = 0; i < 8; ++i) tot += sW[i];
    out[0] = tot / N_PIX;
  }
}

extern "C" void kernel_launch(void* const* d_in, const int* in_sizes, int n_in,
                              void* d_out, int out_size, void* d_ws, size_t ws_size,
                              hipStream_t stream) {
  const float* img1 = (const float*)d_in[0];
  const float* img2 = (const float*)d_in[1];
  const float* gk   = (const float*)d_in[2];
  float* partial = (float*)d_ws;   // N_BLOCKS floats of scratch (~96 KB)
  float* out = (float*)d_out;

  dssim_tile_kernel<<<N_BLOCKS, 128, 0, stream>>>(img1, img2, gk, partial);
  dssim_reduce_kernel<<<1, 256, 0, stream>>>(partial, N_BLOCKS, out);
}